// SaliencyEvaluator_PSRW_7095285973038
// MI455X (gfx1250) — compile-verified
//
#include <hip/hip_runtime.h>
#include <math.h>

namespace {
constexpr int kH = 18, kW = 18, kHW = 324;
constexpr int kB = 256, kC = 324, kBC = kB * kC;   // 82944 rows
constexpr int kQ = 81;                              // float4s per row (1296 B, 16B aligned)
constexpr int kWaves = 8, kThreads = 256;

__device__ __forceinline__ float wsumf(float v) {
#pragma unroll
  for (int m = 16; m; m >>= 1) v += __shfl_xor(v, m, 32);
  return v;
}
__device__ __forceinline__ float wmaxf(float v) {
#pragma unroll
  for (int m = 16; m; m >>= 1) v = fmaxf(v, __shfl_xor(v, m, 32));
  return v;
}
__device__ __forceinline__ int wmini(int v) {
#pragma unroll
  for (int m = 16; m; m >>= 1) { int o = __shfl_xor(v, m, 32); v = (o < v) ? o : v; }
  return v;
}
__device__ __forceinline__ int wsumi(int v) {
#pragma unroll
  for (int m = 16; m; m >>= 1) v += __shfl_xor(v, m, 32);
  return v;
}

// gfx1250 async copy: 16 bytes per active lane, global -> LDS, tracked by ASYNCcnt.
__device__ __forceinline__ void async_g2l_b128(unsigned lds_off, const void* gaddr) {
  asm volatile("global_load_async_to_lds_b128 %0, %1, off"
               :: "v"(lds_off), "v"(gaddr)
               : "memory");
}
__device__ __forceinline__ void wait_asynccnt0() {
  asm volatile("s_wait_asynccnt 0" ::: "memory");
}
}  // namespace

// One wave32 per (b,c) row of the 18x18 cost map.
// Row is staged global->LDS with gfx1250 async loads, then register-resident.
__global__ __launch_bounds__(kThreads)
void psrw_rows_kernel(const float* __restrict__ cv,
                      const int* __restrict__ peaks,
                      float* __restrict__ psrw) {
  __shared__ __align__(16) float smem[kWaves][kHW];   // 1296 B per wave slice

  const int lane = threadIdx.x & 31;
  const int wave = threadIdx.x >> 5;
  const int bc = blockIdx.x * kWaves + wave;

  const float* grow = cv + (size_t)bc * kHW;
  float* wrow = smem[wave];
  const unsigned lds_base = (unsigned)(uintptr_t)wrow;  // low 32 bits = LDS offset

  // Stream hint for the peak coords (gfx1250 global_prefetch_b8).
  __builtin_prefetch(&peaks[2 * bc], 0, 0);

  // Fire the whole row as async copies (3 x 512 B per wave), one counter wait.
  const int q0 = lane, q1 = lane + 32, q2 = lane + 64;
  async_g2l_b128(lds_base + (unsigned)q0 * 16u, grow + q0 * 4);
  async_g2l_b128(lds_base + (unsigned)q1 * 16u, grow + q1 * 4);
  if (q2 < kQ) async_g2l_b128(lds_base + (unsigned)q2 * 16u, grow + q2 * 4);

  const int py = peaks[2 * bc + 0];
  const int px = peaks[2 * bc + 1];

  // Per-element offsets from the peak (clipping never shrinks the minimal
  // offset norm, so scatter-with-clip membership == pure offset test).
  int dy_[12], dx_[12];
  bool ok_[12];
#pragma unroll
  for (int u = 0; u < 12; ++u) {
    const int idx = lane * 4 + (u >> 2) * 128 + (u & 3);
    ok_[u] = (idx < kHW);
    const int y = idx / kW;
    const int x = idx - y * kW;
    dy_[u] = y - py;
    dx_[u] = x - px;
  }

  // Wait for this wave's async copies, then pull the row into registers (ds_load_b128).
  wait_asynccnt0();
  union { float4 q[3]; float f[12]; } vv;
  const float4* w4 = reinterpret_cast<const float4*>(wrow);
  vv.q[0] = w4[q0];
  vv.q[1] = w4[q1];
  vv.q[2] = (q2 < kQ) ? w4[q2] : make_float4(0.f, 0.f, 0.f, 0.f);

  // ---- Pass A: total sum, 3x3-box sum, global max ----
  float s_tot = 0.f, s_rect = 0.f, vmax = -INFINITY;
#pragma unroll
  for (int u = 0; u < 12; ++u) {
    if (ok_[u]) {
      const float xv = vv.f[u];
      s_tot += xv;
      vmax = fmaxf(vmax, xv);
      if (dy_[u] >= -1 && dy_[u] <= 1 && dx_[u] >= -1 && dx_[u] <= 1) s_rect += xv;
    }
  }
  s_tot  = wsumf(s_tot);
  s_rect = wsumf(s_rect);
  vmax   = wmaxf(vmax);

  const int ny = min(py + 1, kH - 1) - max(py - 1, 0) + 1;  // distinct clipped rows
  const int nx = min(px + 1, kW - 1) - max(px - 1, 0) + 1;  // distinct clipped cols
  const float mean_pri = (s_tot - s_rect) / (float)(kHW - ny * nx);

  // ---- Pass B: min distance^2 among kept pixels (cv <= mean_pri, not the peak) ----
  int mind2 = 0x7fffffff;
#pragma unroll
  for (int u = 0; u < 12; ++u) {
    if (ok_[u]) {
      const int d2 = dy_[u] * dy_[u] + dx_[u] * dx_[u];
      if (vv.f[u] <= mean_pri && d2 != 0) mind2 = (d2 < mind2) ? d2 : mind2;
    }
  }
  mind2 = wmini(mind2);
  const float width  = (mind2 == 0x7fffffff) ? 100.0f : sqrtf((float)mind2);
  const float radius = fminf(fmaxf(width, 1.5f), 4.5f);

  // ---- Pass C: masked (disc) sum & count -> mean over the complement ----
  float s_mask = 0.f;
  int   n_mask = 0;
  bool  msk_[12];
#pragma unroll
  for (int u = 0; u < 12; ++u) {
    const int ady = dy_[u] < 0 ? -dy_[u] : dy_[u];
    const int adx = dx_[u] < 0 ? -dx_[u] : dx_[u];
    const int d2  = dy_[u] * dy_[u] + dx_[u] * dx_[u];
    const bool m = ok_[u] && (ady <= 4) && (adx <= 4) &&
                   (sqrtf((float)d2) <= radius);
    msk_[u] = m;
    if (m) { s_mask += vv.f[u]; n_mask += 1; }
  }
  s_mask = wsumf(s_mask);
  n_mask = wsumi(n_mask);

  const float n      = (float)(kHW - n_mask);
  const float mean_s = (s_tot - s_mask) / n;

  // ---- Pass D: variance over the unmasked complement ----
  float s_var = 0.f;
#pragma unroll
  for (int u = 0; u < 12; ++u) {
    if (ok_[u] && !msk_[u]) {
      const float d = vv.f[u] - mean_s;
      s_var += d * d;
    }
  }
  s_var = wsumf(s_var);
  const float var_s = s_var / (n - 1.0f);

  const float p = (vmax - mean_s) / (var_s * width + 1e-16f);
  if (lane == 0) psrw[bc] = p;
}

// Normalize each B-row of psrw by its mean over C. Safe in-place (each block
// owns row b; each thread reads-then-writes only its own elements).
__global__ __launch_bounds__(kThreads)
void psrw_norm_kernel(const float* __restrict__ psrw, float* __restrict__ out) {
  __shared__ float red[kWaves];
  const int b = blockIdx.x;
  const float* row = psrw + (size_t)b * kC;

  float s = 0.f;
  for (int c = threadIdx.x; c < kC; c += kThreads) s += row[c];
  s = wsumf(s);
  if ((threadIdx.x & 31) == 0) red[threadIdx.x >> 5] = s;
  __syncthreads();
  if (threadIdx.x == 0) {
    float t = 0.f;
#pragma unroll
    for (int i = 0; i < kWaves; ++i) t += red[i];
    red[0] = t;
  }
  __syncthreads();
  const float denom = red[0] * (1.0f / (float)kC) + 1e-8f;
  for (int c = threadIdx.x; c < kC; c += kThreads)
    out[(size_t)b * kC + c] = row[c] / denom;
}

extern "C" void kernel_launch(void* const* d_in, const int* in_sizes, int n_in,
                              void* d_out, int out_size, void* d_ws, size_t ws_size,
                              hipStream_t stream) {
  (void)in_sizes; (void)n_in; (void)out_size;
  const float* cv    = (const float*)d_in[0];   // cost_volume  (f32, 256*324*18*18)
  const int*   peaks = (const int*)d_in[1];     // peak_coords  (i32, 256*324*2)
  // d_in[2], d_in[3] (mesh_y, mesh_x) are recomputed analytically — not read.

  float* out = (float*)d_out;
  float* scratch = (ws_size >= (size_t)kBC * sizeof(float)) ? (float*)d_ws : out;

  psrw_rows_kernel<<<kBC / kWaves, kThreads, 0, stream>>>(cv, peaks, scratch);
  psrw_norm_kernel<<<kB, kThreads, 0, stream>>>(scratch, out);
}